// Convolution_59717225284148
// MI455X (gfx1250) — compile-verified
//
#include <hip/hip_runtime.h>
#include <hip/hip_bf16.h>

// ---------------------------------------------------------------------------
// Sparse 2x2x2 conv (stride-2 pattern): each input p hits exactly one anchor
// cell c = p>>1 with tap s = p&1 (3-bit). out[rank(c)] += feat @ K[s].
// Pipeline: zero -> mark cells -> count -> scan -> rank(+positions) -> WMMA
// gather-GEMM-scatter with fp32 V_WMMA_F32_16X16X4_F32.
// ---------------------------------------------------------------------------

typedef float v2f __attribute__((ext_vector_type(2)));
typedef float v8f __attribute__((ext_vector_type(8)));

#define NCELLS   262144      // 64^3
#define NBLK     256         // scan blocks of 1024 cells
#define TILE_ROWS 64         // inputs per workgroup in the GEMM kernel
#define LDSTRIDE  132        // 128 + 4 floats: 16B-aligned rows, 4-bank skew

// ---------------- zero out + workspace ----------------
__global__ void k_zero(float* __restrict__ out, int nOut,
                       unsigned* __restrict__ wsu, int nWs) {
  int i = blockIdx.x * blockDim.x + threadIdx.x;
  int stride = gridDim.x * blockDim.x;
  for (int j = i; j < nOut; j += stride) out[j] = 0.0f;
  for (int j = i; j < nWs; j += stride) wsu[j] = 0u;
}

// ---------------- mark occupied cells ----------------
__global__ void k_mark(const int* __restrict__ pos, unsigned* __restrict__ flags, int N) {
  int i = blockIdx.x * blockDim.x + threadIdx.x;
  if (i < N) {
    int px = pos[3 * i + 0], py = pos[3 * i + 1], pz = pos[3 * i + 2];
    unsigned cid = ((unsigned)(px >> 1) << 12) | ((unsigned)(py >> 1) << 6) |
                   (unsigned)(pz >> 1);
    flags[cid] = 1u;   // benign race: everyone writes 1
  }
}

// ---------------- per-1024-cell block counts ----------------
__global__ __launch_bounds__(256) void k_count(const unsigned* __restrict__ flags,
                                               unsigned* __restrict__ blockSums) {
  __shared__ unsigned red[256];
  int b = blockIdx.x, t = threadIdx.x;
  unsigned s = 0;
  for (int i = t; i < 1024; i += 256) s += flags[b * 1024 + i];
  red[t] = s;
  __syncthreads();
  for (int off = 128; off > 0; off >>= 1) {
    if (t < off) red[t] += red[t + off];
    __syncthreads();
  }
  if (t == 0) blockSums[b] = red[0];
}

// ---------------- exclusive scan of 256 block sums ----------------
__global__ __launch_bounds__(256) void k_scan(const unsigned* __restrict__ blockSums,
                                              unsigned* __restrict__ blockOffs) {
  __shared__ unsigned sc[256];
  int t = threadIdx.x;
  unsigned v = blockSums[t];
  sc[t] = v;
  __syncthreads();
  for (int off = 1; off < 256; off <<= 1) {
    unsigned u = (t >= off) ? sc[t - off] : 0u;
    __syncthreads();
    sc[t] += u;
    __syncthreads();
  }
  blockOffs[t] = sc[t] - v;   // exclusive
}

// ------- ranks in place (flag -> rank | 0xFFFFFFFF) + output positions -------
__global__ __launch_bounds__(1024) void k_rank(unsigned* __restrict__ cellRank,
                                               const unsigned* __restrict__ blockOffs,
                                               float* __restrict__ outPos, int M) {
  __shared__ unsigned sc[1024];
  int t = threadIdx.x;
  int gid = blockIdx.x * 1024 + t;
  unsigned f = cellRank[gid];
  sc[t] = f;
  __syncthreads();
  for (int off = 1; off < 1024; off <<= 1) {
    unsigned u = (t >= off) ? sc[t - off] : 0u;
    __syncthreads();
    sc[t] += u;
    __syncthreads();
  }
  unsigned rank = blockOffs[blockIdx.x] + sc[t] - f;
  if (f) {
    unsigned cx = (unsigned)gid >> 12, cy = ((unsigned)gid >> 6) & 63u,
             cz = (unsigned)gid & 63u;
    if ((int)rank < M) {
      outPos[3 * rank + 0] = (float)cx + 0.25f;   // ((2c)+0.5)/2
      outPos[3 * rank + 1] = (float)cy + 0.25f;
      outPos[3 * rank + 2] = (float)cz + 0.25f;
    }
    cellRank[gid] = rank;
  } else {
    cellRank[gid] = 0xFFFFFFFFu;
  }
}

// ---------------- gather-GEMM-scatter with fp32 WMMA ----------------
// Block: 256 threads = 8 waves; 64 input rows staged in LDS.
// wave w: row-tile (w&3) of 4, column half (w>>2): 8 of 16 col-tiles.
// For each tap d: A rows with s!=d are zero-masked (data mask, EXEC stays all-1).
__global__ __launch_bounds__(256) void k_gemm(const float* __restrict__ feats,
                                              const int* __restrict__ pos,
                                              const float* __restrict__ W,
                                              const unsigned* __restrict__ cellRank,
                                              float* __restrict__ out, int N) {
  __shared__ __align__(16) float As[TILE_ROWS * LDSTRIDE];
  __shared__ unsigned sArr[TILE_ROWS];
  __shared__ unsigned rArr[TILE_ROWS];

  const int tid = threadIdx.x;
  const int row0 = blockIdx.x * TILE_ROWS;

  // --- stage A tile (64 x 128 f32), tap codes, output ranks ---
  {
    int r = tid >> 2;         // 0..63
    int q = tid & 3;          // quarter of a row: 32 floats
    int gi = row0 + r;
    float4* dst = (float4*)&As[r * LDSTRIDE + q * 32];
    if (gi < N) {
      const float4* src = (const float4*)&feats[(size_t)gi * 128 + q * 32];
#pragma unroll
      for (int v = 0; v < 8; ++v) dst[v] = src[v];
    } else {
#pragma unroll
      for (int v = 0; v < 8; ++v) dst[v] = make_float4(0.f, 0.f, 0.f, 0.f);
    }
    if (q == 0) {
      unsigned s = 8u, rk = 0xFFFFFFFFu;
      if (gi < N) {
        int px = pos[3 * gi + 0], py = pos[3 * gi + 1], pz = pos[3 * gi + 2];
        s = (unsigned)(((px & 1) << 2) | ((py & 1) << 1) | (pz & 1));
        unsigned cid = ((unsigned)(px >> 1) << 12) | ((unsigned)(py >> 1) << 6) |
                       (unsigned)(pz >> 1);
        rk = cellRank[cid];
      }
      sArr[r] = s;
      rArr[r] = rk;
    }
  }
  __syncthreads();

  const int wave  = tid >> 5;       // 0..7
  const int lane  = tid & 31;
  const int rt    = wave & 3;       // row tile 0..3
  const int chalf = wave >> 2;      // 0/1 -> col tiles [8*chalf, 8*chalf+8)
  const int m     = lane & 15;
  const int koff  = (lane >> 4) << 1;        // 0 or 2 (half-wave K offset)
  const int rowL  = rt * 16 + m;             // this lane's A row
  const float* Arow = &As[rowL * LDSTRIDE];

  // which taps occur in this row tile (uniform across the wave)
  unsigned tileMask = 0;
#pragma unroll
  for (int i = 0; i < 16; ++i) {
    unsigned s = sArr[rt * 16 + i];
    if (s < 8u) tileMask |= (1u << s);
  }

  for (int ci = 0; ci < 8; ++ci) {
    const int c = chalf * 8 + ci;
    const int n = c * 16 + m;          // output column for B/C fragments
    v8f C = {};
    for (int d = 0; d < 8; ++d) {
      if (!(tileMask & (1u << d))) continue;         // wave-uniform skip
      const float msk = (sArr[rowL] == (unsigned)d) ? 1.0f : 0.0f;
      const float* Bd = &W[(size_t)d * 128 * 256 + n];
#pragma unroll 4
      for (int k4 = 0; k4 < 32; ++k4) {
        const int k = k4 * 4 + koff;
        v2f a, b;
        a.x = Arow[k] * msk;           // A 16x4: lane=M, VGPR0/1 = K,K+1
        a.y = Arow[k + 1] * msk;
        b.x = Bd[(size_t)k * 256];     // B 4x16: lane=N, VGPR0/1 = K,K+1
        b.y = Bd[(size_t)(k + 1) * 256];
        C = __builtin_amdgcn_wmma_f32_16x16x4_f32(false, a, false, b,
                                                  (short)0, C, false, false);
      }
    }
    // scatter C (16x16) into ranked output rows
    const int mbase = (lane >> 4) * 8;   // lanes 16-31 hold M+8
#pragma unroll
    for (int r = 0; r < 8; ++r) {
      const int lrow = rt * 16 + mbase + r;
      const unsigned rk = rArr[lrow];
      if (rk != 0xFFFFFFFFu) {
        atomicAdd(&out[(size_t)rk * 256 + n], C[r]);
      }
    }
  }
}

// ---------------------------------------------------------------------------
extern "C" void kernel_launch(void* const* d_in, const int* in_sizes, int n_in,
                              void* d_out, int out_size, void* d_ws, size_t ws_size,
                              hipStream_t stream) {
  const float* feats = (const float*)d_in[0];     // [N,128] f32
  const int*   pos   = (const int*)d_in[1];       // [N,3]   i32
  const float* W     = (const float*)d_in[2];     // [2,2,2,128,256] f32
  float* out = (float*)d_out;

  const int N = in_sizes[1] / 3;
  const int M = out_size / 259;                   // 256 feat + 3 pos per row

  unsigned* cellRank  = (unsigned*)d_ws;          // NCELLS u32 (flags -> ranks)
  unsigned* blockSums = cellRank + NCELLS;        // NBLK u32
  unsigned* blockOffs = blockSums + NBLK;         // NBLK u32
  const int nWs = NCELLS + 2 * NBLK;

  k_zero<<<512, 256, 0, stream>>>(out, out_size, cellRank, nWs);
  k_mark<<<(N + 255) / 256, 256, 0, stream>>>(pos, cellRank, N);
  k_count<<<NBLK, 256, 0, stream>>>(cellRank, blockSums);
  k_scan<<<1, 256, 0, stream>>>(blockSums, blockOffs);
  k_rank<<<NBLK, 1024, 0, stream>>>(cellRank, blockOffs, out + (size_t)M * 256, M);
  k_gemm<<<(N + TILE_ROWS - 1) / TILE_ROWS, 256, 0, stream>>>(
      feats, pos, W, cellRank, out, N);
}